// CRF_4020089389717
// MI455X (gfx1250) — compile-verified
//
#include <hip/hip_runtime.h>
#include <stdint.h>

constexpr int Bn = 512;
constexpr int Tn = 1024;
constexpr int Ln = 48;
constexpr int BATCH_PER_BLK = 2;
constexpr int THREADS_FWD = BATCH_PER_BLK * Ln;  // 96 = 3 full wave32s

#if __has_builtin(__builtin_amdgcn_global_load_async_to_lds_b32) && \
    __has_builtin(__builtin_amdgcn_s_wait_asynccnt)
#define USE_ASYNC 1
#else
#define USE_ASYNC 0
#endif

#if __has_builtin(__builtin_amdgcn_tensor_load_to_lds) && \
    __has_builtin(__builtin_amdgcn_s_wait_tensorcnt)
#define USE_TDM 1
#else
#define USE_TDM 0
#endif

#if USE_ASYNC
template <int N>
__device__ __forceinline__ void wait_async() {
  __builtin_amdgcn_s_wait_asynccnt(N);
}
#endif

#if USE_TDM
typedef unsigned int u32x4 __attribute__((ext_vector_type(4)));
typedef int          i32x4 __attribute__((ext_vector_type(4)));
typedef int          i32x8 __attribute__((ext_vector_type(8)));
#endif

// ---------------------------------------------------------------------------
// Forward Viterbi: one block = 2 batch rows, thread (b_local, j), j in [0,48).
// Transitions tile DMA'd into LDS by the Tensor Data Mover; v kept in LDS
// (double buffered); emissions streamed via async global->LDS (double
// buffered, ASYNCcnt pipelined); backpointers stored as u8 (L2-resident).
// ---------------------------------------------------------------------------
__global__ __launch_bounds__(THREADS_FWD) void crf_forward(
    const float* __restrict__ emissions,    // [B, T, L]
    const float* __restrict__ transitions,  // [L, L] (row i, col j)
    const float* __restrict__ start_t,      // [L]
    const float* __restrict__ end_t,        // [L]
    uint8_t* __restrict__ bp,               // [T-1, B, L] workspace
    float* __restrict__ vfinal)             // [B, L] workspace
{
  __shared__ __align__(16) float transLds[Ln * Ln];
  __shared__ __align__(16) float vbuf[2][BATCH_PER_BLK][Ln];
#if USE_ASYNC
  __shared__ __align__(16) float embuf[2][BATCH_PER_BLK][Ln];
#endif

  const int tid = threadIdx.x;
  const int bl  = tid / Ln;
  const int j   = tid % Ln;
  const int b   = blockIdx.x * BATCH_PER_BLK + bl;
  const float* emB = emissions + (size_t)b * Tn * Ln + j;  // stride Ln per t

#if USE_TDM
  // One-shot TDM tile fill: 48x48 f32 transitions -> LDS.
  // D# per CDNA5 ISA (group0 128b, group1 256b, 2-D).
  {
    const uint64_t ga   = (uint64_t)(uintptr_t)transitions;
    const uint32_t lofs = (uint32_t)(uintptr_t)&transLds[0];  // LDS byte offset
    u32x4 g0;
    g0.s0 = 1u;                 // count=1 (valid user descriptor), no gather
    g0.s1 = lofs;               // lds_addr [63:32]
    g0.s2 = (uint32_t)ga;       // global_addr[31:0]
    g0.s3 = (uint32_t)((ga >> 32) & 0x1FFFFFFu) | (2u << 30);  // ga[56:32] | type=2
    i32x8 g1;
    g1.s0 = (2 << 16);                      // wg_mask=0, data_size=2 (4 bytes)
    g1.s1 = (Ln & 0xFFFF) << 16;            // tensor_dim0[15:0] = 48
    g1.s2 = ((Ln & 0xFFFF) << 16);          // tensor_dim0 hi=0 | tensor_dim1 lo=48
    g1.s3 = ((Ln & 0xFFFF) << 16);          // tensor_dim1 hi=0 | tile_dim0=48
    g1.s4 = Ln;                             // tile_dim1=48, tile_dim2=0 (2-D)
    g1.s5 = Ln;                             // tensor_dim0_stride = 48 (lo32)
    g1.s6 = 0;                              // stride0 hi | stride1 lo
    g1.s7 = 0;                              // stride1 hi
    i32x4 gz4 = {0, 0, 0, 0};               // groups 2/3 unused (2-D tensor)
    i32x8 gz8 = {0, 0, 0, 0, 0, 0, 0, 0};   // extra group (6-arg form), unused
    __builtin_amdgcn_tensor_load_to_lds(g0, g1, gz4, gz4, gz8, 0);
    __builtin_amdgcn_s_wait_tensorcnt(0);   // TDM done (EXEC-independent issue)
  }
#else
  for (int idx = tid; idx < Ln * Ln; idx += THREADS_FWD)
    transLds[idx] = transitions[idx];
#endif

#if USE_ASYNC
  // Prologue: issue emission rows t=0 and t=1 into the double buffer.
  __builtin_amdgcn_global_load_async_to_lds_b32(
      (int*)(emB + 0 * Ln), (int*)&embuf[0][bl][j], 0, 0);
  __builtin_amdgcn_global_load_async_to_lds_b32(
      (int*)(emB + 1 * Ln), (int*)&embuf[1][bl][j], 0, 0);
  wait_async<1>();  // row 0 landed (per-wave, in-order completion)
  float e0 = embuf[0][bl][j];
#else
  float e0 = emB[0];
#endif
  vbuf[0][bl][j] = start_t[j] + e0;
  __syncthreads();  // trans tile + v0 visible to all 3 waves

  // Hoist transition column T[:, j] into registers (read once per kernel).
  float tcol[Ln];
#pragma unroll
  for (int i = 0; i < Ln; ++i) tcol[i] = transLds[i * Ln + j];

  const float NEG = -3.402823466e38f;
  int vb = 0;
  for (int t = 1; t < Tn; ++t) {
#if USE_ASYNC
    if (t + 1 < Tn) {
      // Issue t+1 into the buffer whose row (t-1) was consumed last iter.
      __builtin_amdgcn_global_load_async_to_lds_b32(
          (int*)(emB + (size_t)(t + 1) * Ln),
          (int*)&embuf[(t + 1) & 1][bl][j], 0, 0);
      wait_async<1>();  // in-order completion => oldest (row t) is done
    } else {
      wait_async<0>();
    }
    float e = embuf[t & 1][bl][j];
#else
    if (t + 1 < Tn) __builtin_prefetch(emB + (size_t)(t + 1) * Ln, 0, 1);
    float e = emB[(size_t)t * Ln];
#endif

    // max-plus inner product over i with 4 independent accumulator chains.
    const float4* vrow = (const float4*)&vbuf[vb][bl][0];
    float b0 = NEG, b1 = NEG, b2 = NEG, b3 = NEG;
    int a0 = 0, a1 = 1, a2 = 2, a3 = 3;
#pragma unroll
    for (int i0 = 0; i0 < Ln; i0 += 4) {
      float4 vv = vrow[i0 >> 2];  // LDS b128 load, broadcast across lanes of same b
      float s0 = vv.x + tcol[i0 + 0];
      float s1 = vv.y + tcol[i0 + 1];
      float s2 = vv.z + tcol[i0 + 2];
      float s3 = vv.w + tcol[i0 + 3];
      if (s0 > b0) { b0 = s0; a0 = i0 + 0; }  // strict '>' keeps first max
      if (s1 > b1) { b1 = s1; a1 = i0 + 1; }
      if (s2 > b2) { b2 = s2; a2 = i0 + 2; }
      if (s3 > b3) { b3 = s3; a3 = i0 + 3; }
    }
    // Combine chains with first-occurrence tie-break (matches jnp.argmax).
    float best = b0; int arg = a0;
    if (b1 > best || (b1 == best && a1 < arg)) { best = b1; arg = a1; }
    if (b2 > best || (b2 == best && a2 < arg)) { best = b2; arg = a2; }
    if (b3 > best || (b3 == best && a3 < arg)) { best = b3; arg = a3; }

    vbuf[vb ^ 1][bl][j] = best + e;
    bp[((size_t)(t - 1) * Bn + b) * Ln + j] = (uint8_t)arg;
    __syncthreads();
    vb ^= 1;
  }

  // Own element only (we wrote it ourselves before the last barrier).
  vfinal[(size_t)b * Ln + j] = vbuf[vb][bl][j] + end_t[j];
}

// ---------------------------------------------------------------------------
// Backtrack: one wave32 per batch. Final argmax via shfl_xor tree, then a
// pointer chase where the 48-byte bp row for step k-1 is loaded (lanes 0..11)
// while step k resolves; the only dependent op per step is one ds_bpermute.
// ---------------------------------------------------------------------------
__global__ __launch_bounds__(32) void crf_backtrack(
    const float* __restrict__ vfinal,  // [B, L]
    const uint8_t* __restrict__ bp,    // [T-1, B, L]
    int* __restrict__ out)             // [B, T]
{
  const int b    = blockIdx.x;
  const int lane = threadIdx.x;

  // argmax over 48 values, first-occurrence tie-break.
  float bv = vfinal[(size_t)b * Ln + lane];
  int   bi = lane;
  if (lane < Ln - 32) {
    float v2 = vfinal[(size_t)b * Ln + lane + 32];
    if (v2 > bv) { bv = v2; bi = lane + 32; }  // strict '>' prefers smaller idx
  }
#pragma unroll
  for (int off = 16; off >= 1; off >>= 1) {
    float ov = __shfl_xor(bv, off, 32);
    int   oi = __shfl_xor(bi, off, 32);
    if (ov > bv || (ov == bv && oi < bi)) { bv = ov; bi = oi; }
  }
  int tag = bi;  // broadcast-consistent across the wave
  if (lane == 0) out[(size_t)b * Tn + (Tn - 1)] = tag;

  // Preload bp row for k = T-2 (12 dwords across lanes 0..11).
  uint32_t cur = 0;
  if (lane < Ln / 4)
    cur = ((const uint32_t*)(bp + ((size_t)(Tn - 2) * Bn + b) * Ln))[lane];

  for (int k = Tn - 2; k >= 0; --k) {
    uint32_t nxt = 0;
    if (k > 0 && lane < Ln / 4)  // chain-independent address -> pipelined load
      nxt = ((const uint32_t*)(bp + ((size_t)(k - 1) * Bn + b) * Ln))[lane];
    if (lane == 0 && k > 4)      // pull upcoming rows toward the WGP cache
      __builtin_prefetch(bp + ((size_t)(k - 4) * Bn + b) * Ln, 0, 1);

    uint32_t w = __shfl(cur, tag >> 2, 32);   // ds_bpermute broadcast
    tag = (int)((w >> ((tag & 3) * 8)) & 0xFFu);
    if (lane == 0) out[(size_t)b * Tn + k] = tag;
    cur = nxt;
  }
}

// ---------------------------------------------------------------------------
extern "C" void kernel_launch(void* const* d_in, const int* in_sizes, int n_in,
                              void* d_out, int out_size, void* d_ws, size_t ws_size,
                              hipStream_t stream) {
  const float* emissions   = (const float*)d_in[0];
  // d_in[1] = mask (unused by the reference decode body)
  const float* transitions = (const float*)d_in[2];
  const float* start_t     = (const float*)d_in[3];
  const float* end_t       = (const float*)d_in[4];

  const size_t bpBytes = (size_t)(Tn - 1) * Bn * Ln;       // ~25.1 MB (u8)
  uint8_t* bp     = (uint8_t*)d_ws;
  float*   vfinal = (float*)((char*)d_ws + bpBytes);       // 16B-aligned

  int* out = (int*)d_out;  // reference output dtype is int32

  crf_forward<<<dim3(Bn / BATCH_PER_BLK), dim3(THREADS_FWD), 0, stream>>>(
      emissions, transitions, start_t, end_t, bp, vfinal);
  crf_backtrack<<<dim3(Bn), dim3(32), 0, stream>>>(vfinal, bp, out);
}